// GraphSAGE_79319456023391
// MI455X (gfx1250) — compile-verified
//
#include <hip/hip_runtime.h>
#include <math.h>

#define N_NODES   100000
#define N_FEAT    128
#define N_CLASSES 64
#define N_EDGES   1600000

typedef __attribute__((ext_vector_type(2))) float v2f;
typedef __attribute__((ext_vector_type(8))) float v8f;

// ---------------- workspace layout (floats) ----------------
// sums: N_NODES * N_FEAT   at offset 0
// deg : N_NODES            after sums
// total: 12,900,000 floats = 51.6 MB

__global__ __launch_bounds__(256) void zero_ws_kernel(float* __restrict__ ws, int n) {
    int i = blockIdx.x * blockDim.x + threadIdx.x;
    if (i < n) ws[i] = 0.0f;
}

// One wave (32 lanes) per edge: gather 128-float row of x[src] as float4/lane,
// scatter-add into sums[dst] with hardware f32 global atomics (L2-resident).
__global__ __launch_bounds__(256) void edge_scatter_kernel(
    const float* __restrict__ x, const int* __restrict__ ei,
    float* __restrict__ sums, float* __restrict__ deg) {
    int wave = (blockIdx.x * blockDim.x + threadIdx.x) >> 5;
    int lane = threadIdx.x & 31;
    if (wave >= N_EDGES) return;
    int src = ei[wave];            // edge_index[0][e]
    int dst = ei[N_EDGES + wave];  // edge_index[1][e]
    const float4* xrow = (const float4*)(x + (size_t)src * N_FEAT);
    float4 v = xrow[lane];
    float* srow = sums + (size_t)dst * N_FEAT + lane * 4;
    unsafeAtomicAdd(srow + 0, v.x);
    unsafeAtomicAdd(srow + 1, v.y);
    unsafeAtomicAdd(srow + 2, v.z);
    unsafeAtomicAdd(srow + 3, v.w);
    if (lane == 0) unsafeAtomicAdd(deg + dst, 1.0f);
}

// ---------------- fused GEMM + bias + log_softmax ----------------
#define BT_STRIDE 65   // 256 x 64 weight-transpose tile, padded (bank-conflict free)
#define A_STRIDE  260  // 16 x 256 A tile, padded (4m+k banks, conflict free)
#define O_STRIDE  68
#define LDS_A_OFF (256 * BT_STRIDE)             // 16640
#define LDS_O_OFF (LDS_A_OFF + 16 * A_STRIDE)   // 20800
#define LDS_FLOATS (LDS_O_OFF + 16 * O_STRIDE)  // 21888 -> 87552 bytes

__global__ __launch_bounds__(128) void sage_gemm_kernel(
    const float* __restrict__ x, const float* __restrict__ sums,
    const float* __restrict__ deg, const float* __restrict__ w_l,
    const float* __restrict__ b_l, const float* __restrict__ w_r,
    float* __restrict__ out) {
    extern __shared__ float smem[];
    float* Bt = smem;                 // Bt[k][c], k in [0,256), c in [0,64)
    float* At = smem + LDS_A_OFF;     // At[r][k], r in [0,16), k in [0,256)
    float* Ot = smem + LDS_O_OFF;     // Ot[r][c]

    const int tid = threadIdx.x;
    const int node0 = blockIdx.x * 16;

    // Stage combined weight matrix transposed: Bt[k][c] = (k<128 ? w_l : w_r)[c][k%128]
    for (int idx = tid; idx < N_CLASSES * N_FEAT; idx += 128) {
        int c = idx >> 7, k = idx & 127;
        Bt[k * BT_STRIDE + c] = w_l[idx];
        Bt[(k + 128) * BT_STRIDE + c] = w_r[idx];
    }
    // Stage A tile: cols [0,128) = mean neighbors, cols [128,256) = x
    for (int idx = tid; idx < 16 * 256; idx += 128) {
        int r = idx >> 8, k = idx & 255;
        int node = node0 + r;
        float v;
        if (k < 128) {
            v = sums[(size_t)node * N_FEAT + k] / fmaxf(deg[node], 1.0f);
        } else {
            v = x[(size_t)node * N_FEAT + (k - 128)];
        }
        At[r * A_STRIDE + k] = v;
    }
    __syncthreads();

    // 4 waves; wave wv computes the 16x16 tile for classes [16*wv, 16*wv+16)
    const int lane = tid & 31;
    const int wv   = tid >> 5;
    const int m    = lane & 15;           // row (A) / col (B) index within tile
    const int koff = (lane >> 4) << 1;    // lanes 16-31 hold K+2 (ISA 16x4 f32 layout)
    const int cb   = wv * 16;

    v8f acc = {};
    const float* arow = At + m * A_STRIDE + koff;
    for (int k = 0; k < 256; k += 4) {
        v2f a, b;
        a.x = arow[k];
        a.y = arow[k + 1];
        b.x = Bt[(k + koff) * BT_STRIDE + cb + m];
        b.y = Bt[(k + koff + 1) * BT_STRIDE + cb + m];
        acc = __builtin_amdgcn_wmma_f32_16x16x4_f32(
            /*neg_a=*/false, a, /*neg_b=*/false, b,
            /*c_mod=*/(short)0, acc, /*reuse_a=*/false, /*reuse_b=*/false);
    }

    // C/D layout: VGPR j -> row (j + 8*hiHalf), col = lane&15
    const int rbase = (lane >> 4) << 3;
#pragma unroll
    for (int j = 0; j < 8; ++j) {
        Ot[(rbase + j) * O_STRIDE + cb + m] = acc[j];
    }
    __syncthreads();

    // log_softmax per node row (threads 0..15, one row each)
    if (tid < 16) {
        int node = node0 + tid;
        float mx = -INFINITY;
        for (int c = 0; c < 64; ++c) {
            float v = Ot[tid * O_STRIDE + c] + b_l[c];
            mx = fmaxf(mx, v);
        }
        float s = 0.0f;
        for (int c = 0; c < 64; ++c) {
            float v = Ot[tid * O_STRIDE + c] + b_l[c];
            s += expf(v - mx);
        }
        float lse = mx + logf(s);
        float* orow = out + (size_t)node * N_CLASSES;
        for (int c = 0; c < 64; ++c) {
            float v = Ot[tid * O_STRIDE + c] + b_l[c];
            orow[c] = v - lse;
        }
    }
}

extern "C" void kernel_launch(void* const* d_in, const int* in_sizes, int n_in,
                              void* d_out, int out_size, void* d_ws, size_t ws_size,
                              hipStream_t stream) {
    const float* x   = (const float*)d_in[0];
    const int*   ei  = (const int*)d_in[1];
    const float* w_l = (const float*)d_in[2];
    const float* b_l = (const float*)d_in[3];
    const float* w_r = (const float*)d_in[4];
    float* out = (float*)d_out;

    float* sums = (float*)d_ws;
    float* deg  = sums + (size_t)N_NODES * N_FEAT;

    const int nz = N_NODES * N_FEAT + N_NODES;
    zero_ws_kernel<<<(nz + 255) / 256, 256, 0, stream>>>((float*)d_ws, nz);

    // one wave per edge, 8 waves per 256-thread block
    const int nblocks = (N_EDGES + 7) / 8;
    edge_scatter_kernel<<<nblocks, 256, 0, stream>>>(x, ei, sums, deg);

    sage_gemm_kernel<<<N_NODES / 16, 128, LDS_FLOATS * sizeof(float), stream>>>(
        x, sums, deg, w_l, b_l, w_r, out);
}